// DifferentiableVectorQuantizer_87033217286406
// MI455X (gfx1250) — compile-verified
//
#include <hip/hip_runtime.h>
#include <math.h>

typedef __attribute__((ext_vector_type(2))) float v2f;
typedef __attribute__((ext_vector_type(8))) float v8f;

#define C_DIM   64
#define K_CODES 2048
#define HW      9216          // 96*96
#define N_IMG   8
#define T_TOK   (N_IMG * HW)  // 73728
#define MBLK    128           // tokens per workgroup (8 waves)
#define NTHR    256
#define KTILE   16
#define NKT     (K_CODES / KTILE)  // 128
#define ZSTR    68            // LDS row stride (floats): 272 B rows -> 16B aligned, 4-bank rotation
#define CSTR    68

// ---------------------------------------------------------------------------
// Kernel 0: codebook squared norms
// ---------------------------------------------------------------------------
__global__ __launch_bounds__(256)
void dvq_cnorm_kernel(const float* __restrict__ cbk, float* __restrict__ cn) {
    int k = blockIdx.x * 256 + threadIdx.x;
    if (k < K_CODES) {
        const float4* p = (const float4*)(cbk + (size_t)k * C_DIM);
        float s = 0.f;
#pragma unroll
        for (int i = 0; i < C_DIM / 4; ++i) {
            float4 v = p[i];
            s += v.x * v.x + v.y * v.y + v.z * v.z + v.w * v.w;
        }
        cn[k] = s;
    }
}

// ---------------------------------------------------------------------------
// Kernel 1: fused WMMA score matmul + running argmax.
// Block = 256 threads (8 waves), owns 128 tokens; wave owns a 16-row stripe.
// Scores s[t][k] = z_t . c_k - 0.5*||c_k||^2  (z-norm constant per row).
// Bias folded into WMMA accumulator init. Codebook tiles double-buffered in
// LDS with register prefetch: one barrier per k-tile, load latency hidden
// behind the 16-WMMA compute of the previous tile.
// ---------------------------------------------------------------------------
__global__ __launch_bounds__(NTHR)
void dvq_argmax_kernel(const float* __restrict__ z,
                       const float* __restrict__ cbk,
                       const float* __restrict__ cn,
                       int* __restrict__ out_idx) {
    __shared__ float zt[MBLK * ZSTR];         // z tile, [token][channel]
    __shared__ float ct[2][KTILE * CSTR];     // codebook tile, double-buffered
    __shared__ float cnl[K_CODES];            // all code norms

    const int tid  = threadIdx.x;
    const int lane = tid & 31;
    const int wave = tid >> 5;
    const int nl   = lane & 15;               // tile column (code) / A row select
    const int hi   = lane >> 4;               // half-wave: selects K pair in frags
    const int t0   = blockIdx.x * MBLK;
    const int n    = t0 / HW;                 // 9216 % 128 == 0 -> no image straddle
    const int hw0  = t0 - n * HW;

    // codebook prefetch slice: one float4 per thread per 4KB tile
    const int kk = tid >> 4;                  // code row in tile (0..15)
    const int c4 = (tid & 15) * 4;            // channel chunk

    // stage code norms
    for (int i = tid; i < K_CODES; i += NTHR) cnl[i] = cn[i];

    // stage z tile transposed: zt[tok][c] = z[n][c][hw0+tok]
    {
        const int tok   = tid & (MBLK - 1);
        const int chalf = tid >> 7;           // 0/1
        const float* zb = z + (size_t)n * C_DIM * HW + hw0 + tok;
#pragma unroll
        for (int i = 0; i < 32; ++i) {
            int c = 2 * i + chalf;
            zt[tok * ZSTR + c] = zb[(size_t)c * HW];
        }
    }

    // stage codebook tile 0, prefetch tile 1 into registers
    {
        float4 v = *(const float4*)(cbk + (size_t)kk * C_DIM + c4);
        *(float4*)&ct[0][kk * CSTR + c4] = v;
    }
    float4 pf = *(const float4*)(cbk + (size_t)(KTILE + kk) * C_DIM + c4);
    __syncthreads();

    float best[8];
    int   bidx[8];
#pragma unroll
    for (int r = 0; r < 8; ++r) { best[r] = -3.4e38f; bidx[r] = 0; }

    // Fragment pointers (conflict-free: bank = 4*row + const over 16 lanes)
    // A (16x4 f32): lane holds row m=nl of the wave stripe, K pair {2*hi, 2*hi+1}
    // B (4x16 f32): lane holds col n=nl, K pair {2*hi, 2*hi+1}
    const float* aPtr = &zt[(wave * 16 + nl) * ZSTR + 2 * hi];
    const float* bP0  = &ct[0][nl * CSTR + 2 * hi];
    const float* bP1  = &ct[1][nl * CSTR + 2 * hi];

    for (int kt = 0; kt < NKT; ++kt) {
        const int kbase = kt * KTILE;
        const float* bPtr = (kt & 1) ? bP1 : bP0;

        // two independent accumulator chains (C split 0..31 / 32..63)
        const float bias = -0.5f * cnl[kbase + nl];
        v8f acc0, acc1;
#pragma unroll
        for (int r = 0; r < 8; ++r) { acc0[r] = bias; acc1[r] = 0.f; }

#pragma unroll
        for (int cb = 0; cb < 32; cb += 4) {
            v2f a0 = *(const v2f*)(aPtr + cb);
            v2f b0 = *(const v2f*)(bPtr + cb);
            acc0 = __builtin_amdgcn_wmma_f32_16x16x4_f32(
                false, a0, false, b0, (short)0, acc0, false, false);
            v2f a1 = *(const v2f*)(aPtr + cb + 32);
            v2f b1 = *(const v2f*)(bPtr + cb + 32);
            acc1 = __builtin_amdgcn_wmma_f32_16x16x4_f32(
                false, a1, false, b1, (short)0, acc1, false, false);
        }

        // D layout: reg r, lanes 0-15 -> row r, lanes 16-31 -> row r+8; col = nl
        const int kidx = kbase + nl;
#pragma unroll
        for (int r = 0; r < 8; ++r) {
            float s = acc0[r] + acc1[r];
            if (s > best[r]) { best[r] = s; bidx[r] = kidx; }  // strict > keeps lowest k-tile
        }

        // publish tile kt+1 (prefetched), start fetching tile kt+2
        if (kt + 1 < NKT) {
            *(float4*)&ct[(kt + 1) & 1][kk * CSTR + c4] = pf;
            if (kt + 2 < NKT)
                pf = *(const float4*)(cbk + (size_t)((kt + 2) * KTILE + kk) * C_DIM + c4);
        }
        __syncthreads();  // retires readers of ct[kt&1], publishes ct[(kt+1)&1]
    }

    // reduce across the 16 columns (xor masks 1..8 stay within each half-wave)
#pragma unroll
    for (int r = 0; r < 8; ++r) {
        float bv = best[r];
        int   bi = bidx[r];
#pragma unroll
        for (int off = 8; off >= 1; off >>= 1) {
            float ov = __shfl_xor(bv, off, 32);
            int   oi = __shfl_xor(bi, off, 32);
            if (ov > bv || (ov == bv && oi < bi)) { bv = ov; bi = oi; }
        }
        if (nl == 0) {
            int m = r + 8 * hi;               // lanes 0 & 16 write rows r and r+8
            out_idx[t0 + wave * 16 + m] = bi;
        }
    }
}

// ---------------------------------------------------------------------------
// Kernel 2: scatter outputs. One thread per (token, channel).
// quantized == hard in forward. Per-token side outputs on c==0.
// ---------------------------------------------------------------------------
__global__ __launch_bounds__(256)
void dvq_scatter_kernel(const float* __restrict__ cbk,
                        const float* __restrict__ ema,
                        const int* __restrict__ idx,
                        float* __restrict__ out) {
    int g = blockIdx.x * 256 + threadIdx.x;
    if (g >= T_TOK * C_DIM) return;
    int t = g >> 6;
    int c = g & (C_DIM - 1);
    int k = idx[t];
    float v = cbk[(size_t)k * C_DIM + c];
    int n  = t / HW;
    int hw = t - n * HW;
    size_t off = (size_t)n * C_DIM * HW + (size_t)c * HW + hw;
    float* q = out;
    float* h = out + (size_t)T_TOK * C_DIM;
    q[off] = v;
    h[off] = v;
    if (c == 0) {
        float* oi = out + 2 * (size_t)T_TOK * C_DIM;
        float* ol = oi + T_TOK;
        float* og = ol + T_TOK;
        float p = fmaxf(ema[k], 1e-9f);
        oi[t] = (float)k;
        ol[t] = p;
        og[t] = logf(p);
    }
}

// ---------------------------------------------------------------------------
// Kernel 3: perplexity = exp(-sum p log p), p = max(ema, eps). K = 2048.
// ---------------------------------------------------------------------------
__global__ __launch_bounds__(256)
void dvq_perplexity_kernel(const float* __restrict__ ema, float* __restrict__ outp) {
    __shared__ float red[256];
    float s = 0.f;
    for (int k = threadIdx.x; k < K_CODES; k += 256) {
        float p = fmaxf(ema[k], 1e-9f);
        s += p * logf(p);
    }
    red[threadIdx.x] = s;
    __syncthreads();
    for (int st = 128; st > 0; st >>= 1) {
        if (threadIdx.x < st) red[threadIdx.x] += red[threadIdx.x + st];
        __syncthreads();
    }
    if (threadIdx.x == 0) outp[0] = expf(-red[0]);
}

// ---------------------------------------------------------------------------
extern "C" void kernel_launch(void* const* d_in, const int* in_sizes, int n_in,
                              void* d_out, int out_size, void* d_ws, size_t ws_size,
                              hipStream_t stream) {
    const float* z   = (const float*)d_in[0];   // [8,64,96,96]
    const float* cbk = (const float*)d_in[1];   // [2048,64]
    const float* ema = (const float*)d_in[2];   // [2048]
    float* out = (float*)d_out;

    // workspace: [0,8KB) code norms; [8KB, 8KB+288KB) token argmin indices
    float* cn  = (float*)d_ws;
    int*   idx = (int*)((char*)d_ws + K_CODES * sizeof(float));

    dvq_cnorm_kernel<<<(K_CODES + 255) / 256, 256, 0, stream>>>(cbk, cn);
    dvq_argmax_kernel<<<T_TOK / MBLK, NTHR, 0, stream>>>(z, cbk, cn, idx);
    dvq_scatter_kernel<<<(T_TOK * C_DIM + 255) / 256, 256, 0, stream>>>(cbk, ema, idx, out);
    float* perp = out + 2 * (size_t)T_TOK * C_DIM + 3 * (size_t)T_TOK;
    dvq_perplexity_kernel<<<1, 256, 0, stream>>>(ema, perp);
}